// RGBMem_42417097016384
// MI455X (gfx1250) — compile-verified
//
#include <hip/hip_runtime.h>

typedef __attribute__((ext_vector_type(2))) float v2f;
typedef __attribute__((ext_vector_type(8))) float v8f;

#define KP1      65537                 // K+1 logits per sample
#define NDIM     128
#define NTILE    4097                  // ceil(KP1/16)
#define LABELS_OFF (16*KP1)            // 1048592
#define NEWMEM_OFF (16*KP1 + 16)       // 1048608
#define INV_T    (1.0f/0.07f)

// ---------------------------------------------------------------------------
// Kernel 1: logits via chained V_WMMA_F32_16X16X4_F32.
// grid.x = b (16), one wave per 16-row tile of gathered memory rows.
// A (16x4 f32 chunk): lane l (l<16) holds row (l&15) elems {4c,4c+1};
//                     lane l+16 holds elems {4c+2,4c+3}          (ISA 7.12.2)
// B (4x16 f32 chunk): same addressing applied to x[col][*] (col = lane&15).
// D column n=b lives in lanes b (M=0..7 in acc[0..7]) and b+16 (M=8..15).
// ---------------------------------------------------------------------------
__global__ __launch_bounds__(256)
void nce_logits_wmma(const float* __restrict__ x,
                     const int*   __restrict__ y,
                     const int*   __restrict__ idx,
                     const float* __restrict__ memory,
                     float*       __restrict__ logits)
{
    const int lane = threadIdx.x & 31;
    const int wave = threadIdx.x >> 5;
    const int b    = blockIdx.x;          // sample 0..15
    const int col  = lane & 15;
    const int half = lane >> 4;           // 0 or 1
    const int off  = 2 * half;

    // Preload B matrix (x, 8 KB, L2-hot) into 64 VGPRs, reused for all tiles.
    v2f bm[32];
    const float* xrow = x + col * NDIM + off;
#pragma unroll
    for (int c = 0; c < 32; ++c)
        bm[c] = *(const v2f*)(xrow + 4 * c);

    const int tstep = gridDim.y * 8;
    for (int tile = blockIdx.y * 8 + wave; tile < NTILE; tile += tstep) {
        const int k0 = tile * 16;
        int k = k0 + col;
        if (k > KP1 - 1) k = KP1 - 1;               // tail tile: clamp (dup gather)
        const int g = (k == 0) ? y[b] : idx[b * KP1 + k];  // positive at k==0
        const float* rowp = memory + (long long)g * NDIM + off;

        v2f a[32];
#pragma unroll
        for (int c = 0; c < 32; ++c)
            a[c] = *(const v2f*)(rowp + 4 * c);     // one full pass over 512B row

        v8f acc = {};
#pragma unroll
        for (int c = 0; c < 32; ++c)
            acc = __builtin_amdgcn_wmma_f32_16x16x4_f32(
                    /*neg_a=*/false, a[c], /*neg_b=*/false, bm[c],
                    /*c_mod=*/(short)0, acc, /*reuse_a=*/false, /*reuse_b=*/false);

        // Extract column n=b of D: lanes b and b+16 hold the 16 logits.
        if (col == b) {
            const int kb = k0 + half * 8;
#pragma unroll
            for (int j = 0; j < 8; ++j) {
                const int kk = kb + j;
                if (kk < KP1) logits[b * KP1 + kk] = acc[j] * INV_T;
            }
        }
    }
}

// ---------------------------------------------------------------------------
// Kernel 2: stream memory -> new_memory with 128-bit vectors (512 MB traffic).
// ---------------------------------------------------------------------------
__global__ __launch_bounds__(256)
void copy_mem(const float4* __restrict__ src, float4* __restrict__ dst, int n4)
{
    const int i = blockIdx.x * 256 + threadIdx.x;
    if (i < n4) dst[i] = src[i];
}

// ---------------------------------------------------------------------------
// Kernel 3: single wave. Writes labels (16 zeros) and overwrites the 16 EMA
// rows deterministically (sequential b-loop => last-write-wins on dup y).
// Each lane owns 4 of the 128 elements; wave32 xor-shuffle reduction.
// ---------------------------------------------------------------------------
__global__ __launch_bounds__(32)
void ema_update(const float* __restrict__ x,
                const int*   __restrict__ y,
                const float* __restrict__ memory,
                float*       __restrict__ out)
{
    const int t = threadIdx.x;            // 0..31
    if (t < 16) out[LABELS_OFF + t] = 0.0f;
    float* newmem = out + NEWMEM_OFF;

    for (int b = 0; b < 16; ++b) {
        const int r = y[b];
        const float4 m4 = ((const float4*)(memory + (long long)r * NDIM))[t];
        const float4 x4 = ((const float4*)(x + b * NDIM))[t];
        float4 v;
        v.x = 0.5f * m4.x + 0.5f * x4.x;
        v.y = 0.5f * m4.y + 0.5f * x4.y;
        v.z = 0.5f * m4.z + 0.5f * x4.z;
        v.w = 0.5f * m4.w + 0.5f * x4.w;
        float ss = v.x * v.x + v.y * v.y + v.z * v.z + v.w * v.w;
#pragma unroll
        for (int o = 16; o > 0; o >>= 1)
            ss += __shfl_xor(ss, o, 32);
        const float inv = 1.0f / fmaxf(sqrtf(ss), 1e-12f);
        float4 w;
        w.x = v.x * inv; w.y = v.y * inv; w.z = v.z * inv; w.w = v.w * inv;
        ((float4*)(newmem + (long long)r * NDIM))[t] = w;
    }
}

// ---------------------------------------------------------------------------
extern "C" void kernel_launch(void* const* d_in, const int* in_sizes, int n_in,
                              void* d_out, int out_size, void* d_ws, size_t ws_size,
                              hipStream_t stream)
{
    const float* x      = (const float*)d_in[0];   // [16,128] f32
    const int*   y      = (const int*)  d_in[1];   // [16] int
    const int*   idx    = (const int*)  d_in[2];   // [16,65537] int
    const float* memory = (const float*)d_in[3];   // [500000,128] f32
    float* out = (float*)d_out;

    // 1) new_memory = memory (256 MB copy, 16M float4s, 62500 blocks exactly)
    const int n4 = (500000 * NDIM) / 4;
    copy_mem<<<(n4 + 255) / 256, 256, 0, stream>>>(
        (const float4*)memory, (float4*)(out + NEWMEM_OFF), n4);

    // 2) logits: grid (b=16, 64 tile-groups) x 256 threads (8 waves/block)
    dim3 grid(16, 64);
    nce_logits_wmma<<<grid, 256, 0, stream>>>(x, y, idx, memory, out);

    // 3) EMA rows + labels (ordered after the copy on the same stream)
    ema_update<<<1, 32, 0, stream>>>(x, y, memory, out);
}